// GCN_21157008900230
// MI455X (gfx1250) — compile-verified
//
#include <hip/hip_runtime.h>

#define N_NODES_C 600000
#define N_EDGES_C 2400000
#define LEADS_C   12
#define NGRAPH_C  (N_NODES_C / LEADS_C)   // 50000
#define HF        64
#define OUTF      32

typedef float v2f __attribute__((ext_vector_type(2)));
typedef float v8f __attribute__((ext_vector_type(8)));

// ---------------------------------------------------------------------------
// Degree / normalization
// ---------------------------------------------------------------------------
__global__ void deg_init(float* __restrict__ deg, int n) {
    int i = blockIdx.x * 256 + threadIdx.x;
    if (i < n) deg[i] = 1.0f;                 // self-loop contributes 1
}

__global__ void deg_count(const int* __restrict__ dst, float* __restrict__ deg, int nE) {
    int e = blockIdx.x * 256 + threadIdx.x;
    if (e < nE) atomicAdd(&deg[dst[e]], 1.0f);
}

__global__ void deg_rsqrt(float* __restrict__ d, int n) {
    int i = blockIdx.x * 256 + threadIdx.x;
    if (i < n) d[i] = rsqrtf(d[i]);           // deg >= 1 always (self loops)
}

// ---------------------------------------------------------------------------
// Node-feature GEMM:  T[n x 64] = H[n x K] @ W[K x 64]  via V_WMMA_F32_16X16X4_F32
// Block = 256 threads = 8 waves; block computes 32 rows x 64 cols
// (2 row-tiles x 4 col-tiles, one 16x16 tile per wave).
// fp32 A 16x4 layout: lane(0..15)=M, VGPR j + lane-half h -> K = 2*h + j.
// B 4x16 mirrored; C/D: VGPR i -> M = i + 8*(lane>=16), N = lane%16.
// ---------------------------------------------------------------------------
template <int K>
__global__ __launch_bounds__(256) void gcn_gemm(const float* __restrict__ H,
                                                const float* __restrict__ W,
                                                float* __restrict__ T, int n) {
    __shared__ __align__(16) float Ws[K * 64];
    __shared__ __align__(16) float As[32 * (K + 4)];   // +4 pad kills bank conflicts
    const int tid = threadIdx.x;
    const int rowBase = blockIdx.x * 32;               // n % 32 == 0

    // stage weights (K*64 floats) and the 32xK A-tile, float4 coalesced
    for (int i = tid * 4; i < K * 64; i += 256 * 4)
        *(float4*)&Ws[i] = *(const float4*)&W[i];
    const int rowChunks = K / 4;
    for (int i = tid; i < 32 * rowChunks; i += 256) {
        int r = i / rowChunks, c = i % rowChunks;
        *(float4*)&As[r * (K + 4) + c * 4] =
            *(const float4*)&H[(size_t)(rowBase + r) * K + c * 4];
    }
    __syncthreads();

    const int wave  = tid >> 5;
    const int lane  = tid & 31;
    const int rtile = wave >> 2;              // 0..1
    const int ctile = wave & 3;               // 0..3
    const int m     = lane & 15;
    const int khalf = (lane >> 4) * 2;        // 0 or 2
    const float* Arow = &As[(rtile * 16 + m) * (K + 4)];
    const int nb = ctile * 16 + m;            // output column for this lane slot

    v8f acc = {};
#pragma unroll
    for (int kk = 0; kk < K; kk += 4) {
        v2f a, b;
        a.x = Arow[kk + khalf + 0];
        a.y = Arow[kk + khalf + 1];
        b.x = Ws[(kk + khalf + 0) * 64 + nb];
        b.y = Ws[(kk + khalf + 1) * 64 + nb];
        acc = __builtin_amdgcn_wmma_f32_16x16x4_f32(false, a, false, b,
                                                    (short)0, acc, false, false);
    }

    const int rOut = rowBase + rtile * 16 + ((lane >> 4) << 3);
#pragma unroll
    for (int i = 0; i < 8; i++)
        T[(size_t)(rOut + i) * 64 + nb] = acc[i];
}

// ---------------------------------------------------------------------------
// Zero accumulator (float4)
// ---------------------------------------------------------------------------
__global__ void zero4(float4* __restrict__ p, int n4) {
    int i = blockIdx.x * 256 + threadIdx.x;
    if (i < n4) p[i] = make_float4(0.f, 0.f, 0.f, 0.f);
}

// ---------------------------------------------------------------------------
// Edge scatter: agg[dst] += t[src] * dinv[src]*dinv[dst].  One wave per edge,
// 2 features per lane (float2 load, 2x global_atomic_add_f32 landing in L2).
// ---------------------------------------------------------------------------
__global__ __launch_bounds__(256) void scatter_edges(const float* __restrict__ t,
                                                     const int* __restrict__ src,
                                                     const int* __restrict__ dst,
                                                     const float* __restrict__ dinv,
                                                     float* __restrict__ agg, int nE) {
    int wave = threadIdx.x >> 5;
    int lane = threadIdx.x & 31;
    int e = blockIdx.x * 8 + wave;
    if (e >= nE) return;
    int s = src[e], d = dst[e];
    float nrm = dinv[s] * dinv[d];
    const float2 tv = *(const float2*)&t[(size_t)s * 64 + lane * 2];
    float* ap = &agg[(size_t)d * 64 + lane * 2];
    atomicAdd(ap + 0, tv.x * nrm);
    atomicAdd(ap + 1, tv.y * nrm);
}

// ---------------------------------------------------------------------------
// Finalize conv layer in place: h = relu(agg + t*dinv^2 + b)  (self-loop fused)
// ---------------------------------------------------------------------------
__global__ void conv_finalize(const float* __restrict__ t,
                              const float* __restrict__ dinv,
                              const float* __restrict__ b,
                              float* __restrict__ h /* agg in, h out */, int n) {
    size_t idx = (size_t)blockIdx.x * 256 + threadIdx.x;   // over n*64
    int i = (int)(idx >> 6);
    int f = (int)(idx & 63);
    if (i >= n) return;
    float di = dinv[i];
    float val = h[idx] + t[idx] * di * di + b[f];
    h[idx] = fmaxf(val, 0.f);
}

// ---------------------------------------------------------------------------
// Fused lead-attention + softmax + (max,mean) pooling + final linear.
// One wave32 per graph: lane owns feature columns {lane, lane+32}.
// ---------------------------------------------------------------------------
__global__ __launch_bounds__(256) void attn_pool(const float* __restrict__ h,
                                                 const float* __restrict__ Wa,
                                                 const float* __restrict__ v,
                                                 const float* __restrict__ Wl,
                                                 const float* __restrict__ bl,
                                                 float* __restrict__ out,
                                                 float* __restrict__ lw_out, int nG) {
    __shared__ float WaS[64 * 64];
    __shared__ float WlS[128 * 32];
    __shared__ float vS[64];
    __shared__ float blS[32];
    for (int i = threadIdx.x; i < 64 * 64; i += 256) WaS[i] = Wa[i];
    for (int i = threadIdx.x; i < 128 * 32; i += 256) WlS[i] = Wl[i];
    if (threadIdx.x < 64) vS[threadIdx.x] = v[threadIdx.x];
    if (threadIdx.x < 32) blS[threadIdx.x] = bl[threadIdx.x];
    __syncthreads();

    const int wave = threadIdx.x >> 5;
    const int lane = threadIdx.x & 31;
    const int g = blockIdx.x * 8 + wave;
    if (g >= nG) return;

    const float* hg = h + (size_t)g * LEADS_C * HF;
    float h0[LEADS_C], h1[LEADS_C];
#pragma unroll
    for (int j = 0; j < LEADS_C; j++) {
        h0[j] = hg[j * HF + lane];
        h1[j] = hg[j * HF + lane + 32];
    }

    // score_j = tanh(h_j @ Wa) . v   (energy computed per lane's 2 columns)
    float score[LEADS_C];
#pragma unroll
    for (int j = 0; j < LEADS_C; j++) {
        float e0 = 0.f, e1 = 0.f;
#pragma unroll
        for (int k = 0; k < 32; k++) {
            float hk = __shfl(h0[j], k, 32);
            e0 += hk * WaS[k * 64 + lane];
            e1 += hk * WaS[k * 64 + lane + 32];
        }
#pragma unroll
        for (int k = 0; k < 32; k++) {
            float hk = __shfl(h1[j], k, 32);
            e0 += hk * WaS[(k + 32) * 64 + lane];
            e1 += hk * WaS[(k + 32) * 64 + lane + 32];
        }
        e0 = tanhf(e0);
        e1 = tanhf(e1);
        float s = e0 * vS[lane] + e1 * vS[lane + 32];
#pragma unroll
        for (int off = 16; off > 0; off >>= 1) s += __shfl_xor(s, off, 32);
        score[j] = s;                          // replicated in all lanes
    }

    // softmax over the 12 leads (replicated per lane)
    float m = score[0];
#pragma unroll
    for (int j = 1; j < LEADS_C; j++) m = fmaxf(m, score[j]);
    float wgt[LEADS_C], sum = 0.f;
#pragma unroll
    for (int j = 0; j < LEADS_C; j++) { wgt[j] = expf(score[j] - m); sum += wgt[j]; }
    float inv = 1.f / sum;
#pragma unroll
    for (int j = 0; j < LEADS_C; j++) wgt[j] *= inv;

    // write lweights [g,12,1]
    float myw = 0.f;
#pragma unroll
    for (int j = 0; j < LEADS_C; j++) myw = (lane == j) ? wgt[j] : myw;
    if (lane < LEADS_C) lw_out[(size_t)g * LEADS_C + lane] = myw;

    // weighted features -> max & mean pool
    float mx0 = -3.402823466e38f, mx1 = -3.402823466e38f, sm0 = 0.f, sm1 = 0.f;
#pragma unroll
    for (int j = 0; j < LEADS_C; j++) {
        float w0 = h0[j] * wgt[j], w1 = h1[j] * wgt[j];
        mx0 = fmaxf(mx0, w0); sm0 += w0;
        mx1 = fmaxf(mx1, w1); sm1 += w1;
    }
    const float p0 = mx0;                // pooled[lane]
    const float p1 = mx1;                // pooled[lane+32]
    const float p2 = sm0 * (1.f / 12.f); // pooled[64+lane]
    const float p3 = sm1 * (1.f / 12.f); // pooled[96+lane]

    // out[o] = relu(sum_p pooled[p]*Wl[p][o] + bl[o]),  o = lane
    float acc = blS[lane];
#pragma unroll
    for (int p = 0; p < 32; p++) acc += __shfl(p0, p, 32) * WlS[p * 32 + lane];
#pragma unroll
    for (int p = 0; p < 32; p++) acc += __shfl(p1, p, 32) * WlS[(32 + p) * 32 + lane];
#pragma unroll
    for (int p = 0; p < 32; p++) acc += __shfl(p2, p, 32) * WlS[(64 + p) * 32 + lane];
#pragma unroll
    for (int p = 0; p < 32; p++) acc += __shfl(p3, p, 32) * WlS[(96 + p) * 32 + lane];
    out[(size_t)g * OUTF + lane] = fmaxf(acc, 0.f);
}

// ---------------------------------------------------------------------------
extern "C" void kernel_launch(void* const* d_in, const int* in_sizes, int n_in,
                              void* d_out, int out_size, void* d_ws, size_t ws_size,
                              hipStream_t stream) {
    const float* x  = (const float*)d_in[0];
    const int*   ei = (const int*)d_in[1];
    // d_in[2] = batch (implicit: node/12), unused
    const float* W1 = (const float*)d_in[3];
    const float* b1 = (const float*)d_in[4];
    const float* W2 = (const float*)d_in[5];
    const float* b2 = (const float*)d_in[6];
    const float* W3 = (const float*)d_in[7];
    const float* b3 = (const float*)d_in[8];
    const float* Wa = (const float*)d_in[9];
    const float* v  = (const float*)d_in[10];
    const float* Wl = (const float*)d_in[11];
    const float* bl = (const float*)d_in[12];

    const int* src = ei;                 // edge_index[0]
    const int* dst = ei + N_EDGES_C;     // edge_index[1]

    float* dinv = (float*)d_ws;                        // [N]
    float* t    = dinv + N_NODES_C;                    // [N,64] transform buffer
    float* hb   = t + (size_t)N_NODES_C * HF;          // [N,64] acc / h (ping)

    float* out = (float*)d_out;                        // [G,32]
    float* lw  = out + (size_t)NGRAPH_C * OUTF;        // [G,12,1]

    const int nbN  = (N_NODES_C + 255) / 256;
    const int nbE  = (N_EDGES_C + 255) / 256;
    const int nbNF4 = (N_NODES_C * HF / 4 + 255) / 256;   // 37500
    const int nbNF  = N_NODES_C * HF / 256;               // 150000

    // --- normalization coefficients ---
    deg_init <<<nbN, 256, 0, stream>>>(dinv, N_NODES_C);
    deg_count<<<nbE, 256, 0, stream>>>(dst, dinv, N_EDGES_C);
    deg_rsqrt<<<nbN, 256, 0, stream>>>(dinv, N_NODES_C);

    // --- layer 1 (K=128) ---
    gcn_gemm<128><<<N_NODES_C / 32, 256, 0, stream>>>(x, W1, t, N_NODES_C);
    zero4<<<nbNF4, 256, 0, stream>>>((float4*)hb, N_NODES_C * HF / 4);
    scatter_edges<<<N_EDGES_C / 8, 256, 0, stream>>>(t, src, dst, dinv, hb, N_EDGES_C);
    conv_finalize<<<nbNF, 256, 0, stream>>>(t, dinv, b1, hb, N_NODES_C);

    // --- layer 2 (K=64) ---
    gcn_gemm<64><<<N_NODES_C / 32, 256, 0, stream>>>(hb, W2, t, N_NODES_C);
    zero4<<<nbNF4, 256, 0, stream>>>((float4*)hb, N_NODES_C * HF / 4);
    scatter_edges<<<N_EDGES_C / 8, 256, 0, stream>>>(t, src, dst, dinv, hb, N_EDGES_C);
    conv_finalize<<<nbNF, 256, 0, stream>>>(t, dinv, b2, hb, N_NODES_C);

    // --- layer 3 (K=64) ---
    gcn_gemm<64><<<N_NODES_C / 32, 256, 0, stream>>>(hb, W3, t, N_NODES_C);
    zero4<<<nbNF4, 256, 0, stream>>>((float4*)hb, N_NODES_C * HF / 4);
    scatter_edges<<<N_EDGES_C / 8, 256, 0, stream>>>(t, src, dst, dinv, hb, N_EDGES_C);
    conv_finalize<<<nbNF, 256, 0, stream>>>(t, dinv, b3, hb, N_NODES_C);

    // --- attention + pooling + classifier ---
    attn_pool<<<NGRAPH_C / 8, 256, 0, stream>>>(hb, Wa, v, Wl, bl, out, lw, NGRAPH_C);
}